// LightGCN_2104533975056
// MI455X (gfx1250) — compile-verified
//
#include <hip/hip_runtime.h>
#include <hip/hip_bf16.h>

#define EMBED 64

typedef __attribute__((ext_vector_type(2))) float v2f;
typedef __attribute__((ext_vector_type(8))) float v8f;

// Guaranteed-native f32 atomic add (global_atomic_add_f32, no CAS loop).
__device__ __forceinline__ void atom_add_f32(float* p, float v) {
#if defined(__HIP_DEVICE_COMPILE__)
    unsafeAtomicAdd(p, v);          // sets unsafe-fp-atomics attr -> native instruction
#else
    *p += v;
#endif
}

// ---------------------------------------------------------------- zero (float4)
__global__ void lg_zero_f4(float4* __restrict__ p, long n4) {
    long i = (long)blockIdx.x * blockDim.x + threadIdx.x;
    if (i < n4) p[i] = make_float4(0.f, 0.f, 0.f, 0.f);
}

// ---------------------------------------------------------------- degree count
__global__ void lg_deg_count(const int* __restrict__ col, float* __restrict__ deg, int E) {
    int e = blockIdx.x * blockDim.x + threadIdx.x;
    if (e < E)
        atom_add_f32(&deg[col[e]], 1.0f);
}

// ---------------------------------------------------------------- deg -> d^{-1/2} in place
__global__ void lg_dinv(float* __restrict__ deg, int N) {
    int n = blockIdx.x * blockDim.x + threadIdx.x;
    if (n < N) {
        float d = deg[n];
        deg[n] = (d > 0.f) ? rsqrtf(d) : 0.f;   // deg>0 => deg>=1, max(deg,1)=deg
    }
}

// ---------------------------------------------------------------- per-edge norm
__global__ void lg_norm(const int* __restrict__ row, const int* __restrict__ col,
                        const float* __restrict__ dinv, float* __restrict__ nrm, int E) {
    int e = blockIdx.x * blockDim.x + threadIdx.x;
    if (e < E) nrm[e] = dinv[row[e]] * dinv[col[e]];
}

// ---------------------------------------------------------------- out = a*x  or  out += a*x
__global__ void lg_axpy(float4* __restrict__ out, const float4* __restrict__ x,
                        const float* __restrict__ alpha, int ai, int init, long n4) {
    long i = (long)blockIdx.x * blockDim.x + threadIdx.x;
    if (i >= n4) return;
    float a = alpha[ai];
    float4 xv = x[i];
    float4 o;
    if (init) {
        o.x = a * xv.x; o.y = a * xv.y; o.z = a * xv.z; o.w = a * xv.w;
    } else {
        o = out[i];
        o.x += a * xv.x; o.y += a * xv.y; o.z += a * xv.z; o.w += a * xv.w;
    }
    out[i] = o;
}

// ---------------------------------------------------------------- scatter: one wave32 per edge
// lane l owns columns {2l, 2l+1}: coalesced 256B gather per row, 2x global_atomic_add_f32
__global__ void lg_scatter(const float* __restrict__ x, const float* __restrict__ nrm,
                           const int* __restrict__ row, const int* __restrict__ col,
                           float* __restrict__ xn, int E) {
    int wave = (int)(((long)blockIdx.x * blockDim.x + threadIdx.x) >> 5);
    int lane = threadIdx.x & 31;
    if (wave >= E) return;
    int r = row[wave];
    int c = col[wave];
    float w = nrm[wave];
    const float2 v = *(const float2*)(x + (long)r * EMBED + lane * 2);
    float* dst = xn + (long)c * EMBED + lane * 2;
    atom_add_f32(dst + 0, v.x * w);
    atom_add_f32(dst + 1, v.y * w);
}

// ---------------------------------------------------------------- pair dots via WMMA f32 16x16x4
// One wave computes 16 pair scores: D = A(src rows 16xK) * B(dst rows^T Kx16),
// accumulated over K=64 in 16 steps of K=4; diagonal of D = desired dots.
// A layout (32-bit, 16x4): lanes 0-15 -> K={k0,k0+1}, lanes 16-31 -> K={k0+2,k0+3}, M = lane&15
// B layout (32-bit, 4x16): same K split across lane halves, N = lane&15
// Diag m: m<8 at (VGPR m, lane m); m>=8 at (VGPR m-8, lane m+16) -> active lanes 0-7, 24-31.
__global__ void lg_pairdot_wmma(const float* __restrict__ out,
                                const int* __restrict__ eli,
                                float* __restrict__ res, int P) {
    int wave = (int)(((long)blockIdx.x * blockDim.x + threadIdx.x) >> 5);
    int lane = threadIdx.x & 31;
    int pbase = wave * 16;
    if (pbase >= P) return;                   // wave-uniform: EXEC stays all-ones for WMMA

    int m  = lane & 15;
    int p  = pbase + m;
    int si = eli[p];                          // edge_label_index[0][p]
    int di = eli[P + p];                      // edge_label_index[1][p]
    const float* srow = out + (long)si * EMBED;
    const float* drow = out + (long)di * EMBED;
    int kOff = (lane >> 4) * 2;

    v8f c = {};
#pragma unroll
    for (int k0 = 0; k0 < EMBED; k0 += 4) {
        v2f a = *(const v2f*)(srow + k0 + kOff);
        v2f b = *(const v2f*)(drow + k0 + kOff);
        c = __builtin_amdgcn_wmma_f32_16x16x4_f32(false, a, false, b,
                                                  (short)0, c, false, false);
    }

    // extract diagonal: VGPR index = lane&7 on active lanes
    int r = lane & 7;
    float val = c[0];
#pragma unroll
    for (int i = 1; i < 8; ++i)
        if (r == i) val = c[i];

    if (lane < 8)        res[pbase + lane]      = val;   // m = 0..7
    else if (lane >= 24) res[pbase + lane - 16] = val;   // m = 8..15
}

// ---------------------------------------------------------------- launcher
extern "C" void kernel_launch(void* const* d_in, const int* in_sizes, int n_in,
                              void* d_out, int out_size, void* d_ws, size_t ws_size,
                              hipStream_t stream) {
    const float* emb   = (const float*)d_in[0];
    const float* alpha = (const float*)d_in[1];
    const int*   ei    = (const int*)d_in[2];
    const int*   eli   = (const int*)d_in[3];

    const int N = in_sizes[0] / EMBED;       // 200000
    const int L = in_sizes[1] - 1;           // 3
    const int E = in_sizes[2] / 2;           // 3200000
    const int P = in_sizes[3] / 2;           // 1000000

    const int* row = ei;                     // (2,E) row-major
    const int* col = ei + E;
    float* res = (float*)d_out;

    // workspace carve-up (all float, multiples of 4 for float4 kernels)
    float* ws  = (float*)d_ws;
    float* deg = ws;                         // N       (becomes dinv in place)
    float* nrm = deg + N;                    // E
    const long nf = (long)N * EMBED;         // 12.8M
    float* xa  = nrm + E;                    // nf
    float* xb  = xa + nf;                    // nf
    float* ov  = xb + nf;                    // nf  -> total ~167 MB

    const int B = 256;
    const long n4 = nf / 4;

    // symmetric normalization
    lg_zero_f4<<<(int)((N / 4 + B - 1) / B), B, 0, stream>>>((float4*)deg, N / 4);
    lg_deg_count<<<(E + B - 1) / B, B, 0, stream>>>(col, deg, E);
    lg_dinv<<<(N + B - 1) / B, B, 0, stream>>>(deg, N);
    lg_norm<<<(E + B - 1) / B, B, 0, stream>>>(row, col, deg, nrm, E);

    // out = alpha[0] * emb
    lg_axpy<<<(int)((n4 + B - 1) / B), B, 0, stream>>>((float4*)ov, (const float4*)emb,
                                                       alpha, 0, 1, n4);

    // 3 rounds of propagation, ping-pong xa/xb, layer-0 reads emb directly
    const float* xcur = emb;
    float* bufs[2] = { xa, xb };
    const long scatterThreads = (long)E * 32;
    const int scatterBlocks = (int)((scatterThreads + B - 1) / B);
    for (int l = 0; l < L; ++l) {
        float* xnext = bufs[l & 1];
        lg_zero_f4<<<(int)((n4 + B - 1) / B), B, 0, stream>>>((float4*)xnext, n4);
        lg_scatter<<<scatterBlocks, B, 0, stream>>>(xcur, nrm, row, col, xnext, E);
        lg_axpy<<<(int)((n4 + B - 1) / B), B, 0, stream>>>((float4*)ov, (const float4*)xnext,
                                                           alpha, l + 1, 0, n4);
        xcur = xnext;
    }

    // pair dot products via WMMA (16 pairs per wave)
    const long nWaves = (P + 15) / 16;
    const long pdThreads = nWaves * 32;
    lg_pairdot_wmma<<<(int)((pdThreads + B - 1) / B), B, 0, stream>>>(ov, eli, res, P);
}